// GCN_Layer_25812753448978
// MI455X (gfx1250) — compile-verified
//
#include <hip/hip_runtime.h>

typedef __attribute__((ext_vector_type(2))) float v2f;
typedef __attribute__((ext_vector_type(8))) float v8f;

#define HN   512
#define WN   512
#define CC   64
#define HWN  (HN * WN)

#define TILE 16
#define HALO 18                    // TILE + 2
#define NPIX (HALO * HALO)         // 324 halo pixels
#define NPG  ((NPIX + 15) / 16)    // 21 pixel groups of 16
#define PSTR 33                    // dwords (fp16 pairs) per pixel in LDS; 33 coprime w/ 64 banks

static __device__ __forceinline__ unsigned int pack2h(float a, float b) {
    union { _Float16 h[2]; unsigned int u; } u;
    u.h[0] = (_Float16)a;
    u.h[1] = (_Float16)b;
    return u.u;
}

static __device__ __forceinline__ float dinv_at(int h, int w) {
    float dh = 3.0f - (float)(h == 0) - (float)(h == HN - 1);
    float dw = 3.0f - (float)(w == 0) - (float)(w == WN - 1);
    return __frsqrt_rn(dh * dw);
}

__global__ __launch_bounds__(256)
void gcn_fused_wmma_kernel(const float* __restrict__ x,      // [B,64,512,512]
                           const float* __restrict__ wgt,    // [64,64] (c,o)
                           float* __restrict__ out)          // [B,64,512,512]
{
    // dinv-scaled xw staged as packed fp16 pairs: [pixel][chan-pair], padded stride.
    __shared__ unsigned int s_xw[NPIX * PSTR];   // 324*33*4 = 42768 bytes

    const int tid  = threadIdx.x;
    const int lane = tid & 31;
    const int wv   = tid >> 5;          // wave id 0..7
    const int l16  = lane & 15;         // M/N position within 16
    const int lh   = lane >> 4;         // lane half -> K offset (+2) / M offset (+8)

    const int blk = blockIdx.x;         // 4096 = B * 32 * 32
    const int b   = blk >> 10;
    const int ty  = (blk >> 5) & 31;
    const int tx  = blk & 31;
    const int h0  = ty * TILE;
    const int w0  = tx * TILE;

    const float* xb = x + (size_t)b * CC * HWN;   // wave-uniform base

    // ---------------- Phase 1: per-pixel 64x64 matmul via WMMA ----------------
    // D[o][pix] = sum_c Wt[o][c] * x[c][pix];  A = Wt (M=o,K=c), B = x (K=c,N=pix)
    // Wave wv owns output-channel group og = wv&3 (16 channels); A stays resident.
    const int og = wv & 3;

    v2f Areg[16];
    {
        const int o = og * 16 + l16;              // A: lanes 0-15 hold the M rows
        #pragma unroll
        for (int kk = 0; kk < 16; ++kk) {
            const int c0 = kk * 4 + 2 * lh;       // VGPR0: K=k0 / k0+2 ; VGPR1: +1
            Areg[kk].x = wgt[c0 * CC + o];
            Areg[kk].y = wgt[(c0 + 1) * CC + o];
        }
    }

    for (int pg = (wv >> 2); pg < NPG; pg += 2) {
        const int p      = pg * 16 + l16;         // this lane's halo pixel (N index)
        const bool vst   = (p < NPIX);
        const int  pp    = vst ? p : 0;
        const int  hh    = pp / HALO;
        const int  ww    = pp - hh * HALO;
        const int  gh    = h0 - 1 + hh;
        const int  gw    = w0 - 1 + ww;
        const bool vpx   = (gh >= 0) && (gh < HN) && (gw >= 0) && (gw < WN);

        // Clamped, always-in-bounds element offset; OOB lanes are zeroed at the
        // store below (via di mask), NOT at the loads -> loads are unconditional.
        const unsigned po  = vpx ? (unsigned)(gh * WN + gw) : 0u;
        const unsigned idx = po + (unsigned)(2 * lh * HWN); // per-lane 32-bit offset

        // Batch all 32 B loads (maximize outstanding loads), then WMMA chain.
        v2f Breg[16];
        #pragma unroll
        for (int kk = 0; kk < 16; ++kk) {
            const float* xb_k = xb + (size_t)(kk * 4) * HWN;  // uniform -> SGPR base
            Breg[kk].x = xb_k[idx];
            Breg[kk].y = xb_k[idx + HWN];                     // +1MB imm offset
        }

        v8f acc = {};
        #pragma unroll
        for (int kk = 0; kk < 16; ++kk) {
            acc = __builtin_amdgcn_wmma_f32_16x16x4_f32(
                      false, Areg[kk], false, Breg[kk], (short)0, acc, false, false);
        }

        // Scale by dinv(pixel); mask OOB halo pixels to exactly zero here.
        const float di = vpx ? dinv_at(gh, gw) : 0.0f;
        if (vst) {
            const int ob = og * 16 + 8 * lh;      // channel base for this lane's D rows
            unsigned int* dst = &s_xw[p * PSTR + (ob >> 1)];
            #pragma unroll
            for (int v = 0; v < 8; v += 2)
                dst[v >> 1] = pack2h(acc[v] * di, acc[v + 1] * di);
        }
    }

    __syncthreads();

    // ---------------- Phase 2: 3x3 box stencil + dinv, write NCHW ----------------
    // Wave wv handles channel pairs [wv*4, wv*4+4). Lanes run along w (coalesced).
    #pragma unroll
    for (int q = 0; q < 4; ++q) {
        const int cp = wv * 4 + q;                // pair index 0..31
        const int c  = cp * 2;
        #pragma unroll 2
        for (int r0 = 0; r0 < TILE; r0 += 2) {
            const int r  = r0 + lh;
            const int wc = l16;
            float s0 = 0.0f, s1 = 0.0f;
            #pragma unroll
            for (int dr = 0; dr < 3; ++dr) {
                #pragma unroll
                for (int dw = 0; dw < 3; ++dw) {
                    const unsigned int pk =
                        s_xw[((r + dr) * HALO + (wc + dw)) * PSTR + cp];
                    union { unsigned int u; _Float16 h[2]; } u;
                    u.u = pk;
                    s0 += (float)u.h[0];
                    s1 += (float)u.h[1];
                }
            }
            const int gh = h0 + r;
            const int gw = w0 + wc;
            const float di = dinv_at(gh, gw);
            const size_t oidx = (((size_t)b * CC + c) * HN + gh) * WN + gw;
            out[oidx]               = s0 * di;
            out[oidx + (size_t)HWN] = s1 * di;
        }
    }
}

extern "C" void kernel_launch(void* const* d_in, const int* in_sizes, int n_in,
                              void* d_out, int out_size, void* d_ws, size_t ws_size,
                              hipStream_t stream) {
    const float* x   = (const float*)d_in[0];   // [4,64,512,512] fp32
    const float* wgt = (const float*)d_in[1];   // [1,64,64] fp32
    float* out       = (float*)d_out;           // [4,64,512,512] fp32

    const int num_blocks = 4 * (HN / TILE) * (WN / TILE);   // 4096
    gcn_fused_wmma_kernel<<<num_blocks, 256, 0, stream>>>(x, wgt, out);
}